// RotationPerturbationLayer_7464653160916
// MI455X (gfx1250) — compile-verified
//
#include <hip/hip_runtime.h>
#include <hip/hip_bf16.h>

typedef __attribute__((ext_vector_type(2))) float v2f;
typedef __attribute__((ext_vector_type(8))) float v8f;

namespace {
constexpr int   kC = 3, kH = 80, kW = 80, kN = kH * kW;   // N = 6400
constexpr float kCx = (kW - 1) * 0.5f;
constexpr float kCy = (kH - 1) * 0.5f;
constexpr int   kImgFloats = kC * kH * kW;                // 19200
constexpr int   kImgVecs   = kImgFloats * 4 / 16;         // 4800 x 16B = 76.8 KB

__device__ __forceinline__ float tentf(float o) {
  // max(0, 1-|o|) written exactly as the reference: 0.5*(z+|z|)
  float z = 1.0f - fabsf(o);
  return 0.5f * (z + fabsf(z));
}
} // namespace

__global__ __launch_bounds__(256, 1)
void rot_tent_wmma(const float* __restrict__ theta,
                   const float* __restrict__ image,
                   float* __restrict__ out) {
  // Whole image [c][y][x] staged once in LDS (76.8 KB of the WGP's 320 KB).
  __shared__ float simg[kImgFloats];  // single shared object -> LDS offset 0

  const int tid  = threadIdx.x;
  const int lane = tid & 31;
  const int wave = tid >> 5;
  const int lo   = lane & 15;      // WMMA col index / A row index
  const int hi   = lane >> 4;      // lane-half: selects K pair & D row block

  const int b      = blockIdx.y;               // angle index, 0..3
  const int tile   = blockIdx.x * 8 + wave;    // 16-row n-tile, 0..399 (exact)
  const int n_base = tile * 16;

  // ---- Single async DMA burst: image -> LDS (CDNA5 async-to-LDS path) ----
  // NOTE: &simg[0] is passed as an asm operand (unused in the template) so the
  // LDS object ESCAPES: alias analysis must then assume the asm writes it,
  // keeping the later ds_loads alive and ordered after the DMA + wait.
  const float* lds_escape = &simg[0];
  for (int i = tid; i < kImgVecs; i += 256) {
    unsigned off = (unsigned)i * 16u;          // same byte offset in LDS & global
    asm volatile("global_load_async_to_lds_b128 %0, %1, %2"
                 :: "v"(off), "v"(off), "s"(image), "s"(lds_escape)
                 : "memory");
  }
  asm volatile("s_wait_asynccnt 0" ::: "memory");

  // ---- Per-lane rotation geometry (computed after the DMA wait) ----
  const float th  = theta[b] * 0.017453292519943295f;  // deg2rad
  const float cth = cosf(th);
  const float sth = sinf(th);

  float src_x_m;                    // A-matrix row m = lo
  {
    int   n  = n_base + lo;
    float gx = (float)(n % kW) - kCx;
    float gy = (float)(n / kW) - kCy;
    src_x_m  = cth * gx + sth * gy + kCx;
  }
  float srcy[8];                    // D-matrix rows v + 8*hi
  #pragma unroll
  for (int v = 0; v < 8; ++v) {
    int   n  = n_base + v + 8 * hi;
    float gx = (float)(n % kW) - kCx;
    float gy = (float)(n / kW) - kCy;
    srcy[v]  = -sth * gx + cth * gy + kCy;
  }

  // A tiles: wx tent weights. Invariant across channels/column-tiles.
  // Lane holds row m=lo, K = 2*hi + j.
  v2f aarr[20];
  #pragma unroll
  for (int kt = 0; kt < 20; ++kt) {
    float xk = (float)(kt * 4 + 2 * hi);
    aarr[kt].x = tentf(src_x_m - xk);
    aarr[kt].y = tentf(src_x_m - (xk + 1.0f));
  }

  __syncthreads();  // LDS image visible to all waves; never overwritten

  for (int c = 0; c < kC; ++c) {
    // Stage 1 (FP32 WMMA): tmp[n, y] = sum_x wx[n,x] * img[c,y,x]
    // fused with Stage 2: pacc += wy[n,y] * tmp[n,y] over 5 y-tiles.
    v8f pacc = {};
    for (int cty = 0; cty < 5; ++cty) {
      const int    ycol = cty * 16 + lo;       // this lane's column -> image row y
      const float* rowp = &simg[(c * kH + ycol) * kW];

      // B tiles: lane holds col=lo (y=ycol), K = 2*hi + j  -> ds_load_b64 x20
      v2f bv[20];
      #pragma unroll
      for (int kt = 0; kt < 20; ++kt)
        bv[kt] = *(const v2f*)(rowp + kt * 4 + 2 * hi);

      v8f acc0 = {}, acc1 = {};
      #pragma unroll
      for (int kt = 0; kt < 20; kt += 2) {
        acc0 = __builtin_amdgcn_wmma_f32_16x16x4_f32(
                   false, aarr[kt],     false, bv[kt],     (short)0, acc0, false, false);
        acc1 = __builtin_amdgcn_wmma_f32_16x16x4_f32(
                   false, aarr[kt + 1], false, bv[kt + 1], (short)0, acc1, false, false);
      }

      // fold wy weight (D row = v + 8*hi, col y = ycol)
      #pragma unroll
      for (int v = 0; v < 8; ++v)
        pacc[v] += (acc0[v] + acc1[v]) * tentf(srcy[v] - (float)ycol);
    }

    // Reduce the 16 columns held across each half-wave (wave32 butterfly).
    #pragma unroll
    for (int off = 1; off < 16; off <<= 1) {
      #pragma unroll
      for (int v = 0; v < 8; ++v)
        pacc[v] += __shfl_xor(pacc[v], off, 32);
    }

    if (lo == 0) {
      float* op = out + ((size_t)b * kC + c) * kN + n_base + 8 * hi;
      #pragma unroll
      for (int v = 0; v < 8; ++v) op[v] = pacc[v];
    }
  }
}

extern "C" void kernel_launch(void* const* d_in, const int* in_sizes, int n_in,
                              void* d_out, int out_size, void* d_ws, size_t ws_size,
                              hipStream_t stream) {
  const float* theta = (const float*)d_in[0];   // [4]  f32, degrees
  const float* image = (const float*)d_in[1];   // [3,80,80] f32
  float*       out   = (float*)d_out;           // [4,3,80,80] f32

  dim3 grid(50, 4, 1);   // 50*8 waves = 400 n-tiles, y = batch angle
  dim3 block(256, 1, 1); // 8 waves (wave32)
  rot_tent_wmma<<<grid, block, 0, stream>>>(theta, image, out);

  (void)in_sizes; (void)n_in; (void)out_size; (void)d_ws; (void)ws_size;
}